// HCINA_MODEL_13280038880161
// MI455X (gfx1250) — compile-verified
//
#include <hip/hip_runtime.h>

#define F_IN 256
#define DOUT 128
#define NEDGE 50000

typedef __attribute__((ext_vector_type(2))) float v2f;
typedef __attribute__((ext_vector_type(8))) float v8f;

// Guaranteed hardware fp32 atomic-add at device scope (L2 atomic units),
// non-returning form (STOREcnt): avoids any chance of a CAS-loop fallback.
__device__ __forceinline__ void atomAddF32(float* p, float v) {
  asm volatile("global_atomic_add_f32 %0, %1, off scope:SCOPE_DEV"
               : : "v"(p), "v"(v) : "memory");
}

// ---------------------------------------------------------------------------
// GEMM: out[N,128] = A[N,256] @ W[256,128] + bias, via V_WMMA_F32_16X16X4_F32.
// One wave per 16x16 output tile, 64 WMMA steps over K=256.
// ---------------------------------------------------------------------------
__global__ void __launch_bounds__(256) hgnn_gemm_wmma(
    const float* __restrict__ A, const float* __restrict__ W,
    const float* __restrict__ bias, float* __restrict__ out, int nrows)
{
  const int lane = threadIdx.x & 31;
  const int wave = threadIdx.x >> 5;
  const long tile = (long)blockIdx.x * 8 + wave;
  const long ntiles = (long)(nrows / 16) * (DOUT / 16);
  if (tile >= ntiles) return;                 // wave-uniform: EXEC stays all-1s
  const int tn = (int)(tile & (DOUT / 16 - 1));
  const int tm = (int)(tile / (DOUT / 16));
  const int half = lane >> 4;                 // 0: K pair (k,k+1), 1: (k+2,k+3)
  const int l15 = lane & 15;

  const float* arow = A + (size_t)(tm * 16 + l15) * F_IN;
  const float* bcol = W + (tn * 16 + l15);

  v8f c = {};
  for (int k = 0; k < F_IN; k += 4) {
    const int ka = k + half * 2;
    v2f a  = { arow[ka], arow[ka + 1] };                       // b64 load
    v2f bm = { bcol[(size_t)ka * DOUT], bcol[(size_t)(ka + 1) * DOUT] };
    c = __builtin_amdgcn_wmma_f32_16x16x4_f32(
        /*neg_a=*/false, a, /*neg_b=*/false, bm,
        /*c_mod=*/(short)0, c, /*reuse_a=*/false, /*reuse_b=*/false);
  }

  const float bv = bias[tn * 16 + l15];
  float* orow = out + (size_t)(tm * 16 + half * 8) * DOUT + tn * 16 + l15;
#pragma unroll
  for (int r = 0; r < 8; ++r)
    orow[(size_t)r * DOUT] = c[r] + bv;
}

// ---------------------------------------------------------------------------
// Weighted scatter: one wave per incidence pair. Each lane handles 4 floats
// (float4 gather = global_load_b128, 4x global_atomic_add_f32 scatter).
// Optional src_den fuses the previous pass's mean-normalization into the
// gather: scale = w / max(den[src], eps). Lane 0 accumulates the denominator.
// Pair index made wave-uniform via readfirstlane so the index/weight fetches
// scalarize (SMEM) instead of 32 redundant vector lane loads.
// ---------------------------------------------------------------------------
__global__ void __launch_bounds__(256) hgnn_scatter_wmean(
    const float* __restrict__ src, const float* __restrict__ src_den,
    const int* __restrict__ sidx, const int* __restrict__ didx,
    const float* __restrict__ wgt,
    float* __restrict__ dst_num, float* __restrict__ dst_den, int nnz)
{
  const int p = __builtin_amdgcn_readfirstlane(
      (int)((blockIdx.x * blockDim.x + threadIdx.x) >> 5));
  const int lane = threadIdx.x & 31;
  if (p >= nnz) return;

  const int s = sidx[p];
  const int d = didx[p];
  const float w = wgt[p];
  float scale = w;
  if (src_den) scale *= 1.0f / fmaxf(src_den[s], 1e-12f);

  const float4 v = ((const float4*)(src + (size_t)s * DOUT))[lane];
  float* dp = dst_num + (size_t)d * DOUT + lane * 4;
  atomAddF32(dp + 0, v.x * scale);
  atomAddF32(dp + 1, v.y * scale);
  atomAddF32(dp + 2, v.z * scale);
  atomAddF32(dp + 3, v.w * scale);
  if (lane == 0) atomAddF32(dst_den + d, w);
}

// ---------------------------------------------------------------------------
// Fused row-normalize (num/den) + row softmax. One wave per row of 128,
// 4 elems per lane, wave32 shfl_xor reductions for max and sum.
// ---------------------------------------------------------------------------
__global__ void __launch_bounds__(256) hgnn_softmax_div(
    const float* __restrict__ num, const float* __restrict__ den,
    float* __restrict__ out, int rows)
{
  const int r = (int)((blockIdx.x * blockDim.x + threadIdx.x) >> 5);
  const int lane = threadIdx.x & 31;
  if (r >= rows) return;

  const float inv = 1.0f / fmaxf(den[r], 1e-12f);
  float4 v = ((const float4*)(num + (size_t)r * DOUT))[lane];
  v.x *= inv; v.y *= inv; v.z *= inv; v.w *= inv;

  float m = fmaxf(fmaxf(v.x, v.y), fmaxf(v.z, v.w));
#pragma unroll
  for (int off = 16; off > 0; off >>= 1)
    m = fmaxf(m, __shfl_xor(m, off, 32));

  float4 e;
  e.x = __expf(v.x - m); e.y = __expf(v.y - m);
  e.z = __expf(v.z - m); e.w = __expf(v.w - m);
  float sum = e.x + e.y + e.z + e.w;
#pragma unroll
  for (int off = 16; off > 0; off >>= 1)
    sum += __shfl_xor(sum, off, 32);

  const float rs = 1.0f / sum;
  float4 o = { e.x * rs, e.y * rs, e.z * rs, e.w * rs };
  ((float4*)(out + (size_t)r * DOUT))[lane] = o;
}

// ---------------------------------------------------------------------------
// Host-side orchestration (graph-capture safe: only async ops on `stream`).
// ---------------------------------------------------------------------------
extern "C" void kernel_launch(void* const* d_in, const int* in_sizes, int n_in,
                              void* d_out, int out_size, void* d_ws, size_t ws_size,
                              hipStream_t stream) {
  const float* feats  = (const float*)d_in[0];
  const float* W      = (const float*)d_in[1];
  const float* bias   = (const float*)d_in[2];
  const float* v2e_w  = (const float*)d_in[3];
  const float* e2v_w  = (const float*)d_in[4];
  const int*   pair_v = (const int*)d_in[5];
  const int*   pair_e = (const int*)d_in[6];

  const int N   = in_sizes[0] / F_IN;   // 200000
  const int NNZ = in_sizes[3];          // 2000000
  const int E   = NEDGE;                // 50000

  // Workspace layout: xA[N*128] | y[E*128] | denE[E] | denN[N]
  float* xA   = (float*)d_ws;
  float* y    = xA + (size_t)N * DOUT;
  float* denE = y + (size_t)E * DOUT;
  float* denN = denE + E;
  float* xB   = (float*)d_out;          // second vertex buffer between hops

  const size_t yBytes  = ((size_t)E * DOUT + E) * sizeof(float); // y + denE
  const size_t xBytes  = (size_t)N * DOUT * sizeof(float);
  const size_t dnBytes = (size_t)N * sizeof(float);

  const int ntiles     = (N / 16) * (DOUT / 16);
  const int gemmBlocks = (ntiles + 7) / 8;      // 8 waves/block
  const int scatBlocks = (NNZ + 7) / 8;         // 1 wave per pair
  const int rowBlocksN = (N + 7) / 8;           // 1 wave per row

  // x = feats @ W + b
  hgnn_gemm_wmma<<<gemmBlocks, 256, 0, stream>>>(feats, W, bias, xA, N);

  // hop 1: v2e
  hipMemsetAsync(y, 0, yBytes, stream);
  hgnn_scatter_wmean<<<scatBlocks, 256, 0, stream>>>(
      xA, nullptr, pair_v, pair_e, v2e_w, y, denE, NNZ);
  // hop 1: e2v (gather fuses y /= denE)
  hipMemsetAsync(xB, 0, xBytes, stream);
  hipMemsetAsync(denN, 0, dnBytes, stream);
  hgnn_scatter_wmean<<<scatBlocks, 256, 0, stream>>>(
      y, denE, pair_e, pair_v, e2v_w, xB, denN, NNZ);

  // hop 2: v2e (gather fuses xB /= denN)
  hipMemsetAsync(y, 0, yBytes, stream);
  hgnn_scatter_wmean<<<scatBlocks, 256, 0, stream>>>(
      xB, denN, pair_v, pair_e, v2e_w, y, denE, NNZ);
  // hop 2: e2v
  hipMemsetAsync(xA, 0, xBytes, stream);
  hipMemsetAsync(denN, 0, dnBytes, stream);
  hgnn_scatter_wmean<<<scatBlocks, 256, 0, stream>>>(
      y, denE, pair_e, pair_v, e2v_w, xA, denN, NNZ);

  // final normalize + softmax -> d_out
  hgnn_softmax_div<<<rowBlocksN, 256, 0, stream>>>(xA, denN, xB, N);
}